// PABV2_65541200937402
// MI455X (gfx1250) — compile-verified
//
#include <hip/hip_runtime.h>

typedef __attribute__((ext_vector_type(2))) float v2f;
typedef __attribute__((ext_vector_type(8))) float v8f;

#define B_   4
#define C_   32
#define H_   128
#define W_   256
#define CR_  85            // W // 3
#define EPS_ 1e-5f
#define XSTR 34            // padded row stride (floats) for [W][C] LDS arrays
#define WSTR 34            // padded stride for effective-weight rows [C][C]

// LDS partition (float offsets)
#define OFF_XQ 0
#define OFF_XK (OFF_XQ + W_*XSTR)
#define OFF_Q  (OFF_XK + W_*XSTR)
#define OFF_K  (OFF_Q  + W_*XSTR)
#define OFF_WQ (OFF_K  + W_*XSTR)
#define OFF_WK (OFF_WQ + C_*WSTR)
#define OFF_BQ (OFF_WK + C_*WSTR)
#define OFF_BK (OFF_BQ + C_)
#define LDS_FLOATS (OFF_BK + C_)

// ---------------------------------------------------------------------------
// out0 = 2*x_left, out1 = 2*x_right  (pure streaming, float4)
// ---------------------------------------------------------------------------
__global__ void pab_double_kernel(const float* __restrict__ xl,
                                  const float* __restrict__ xr,
                                  float* __restrict__ out, int n1) {
  int i = blockIdx.x * blockDim.x + threadIdx.x;   // float4 index over both tensors
  int q = n1 >> 2;                                 // float4s per tensor
  const float4* src = (i < q) ? (const float4*)xl : (const float4*)xr;
  int idx = (i < q) ? i : (i - q);
  float4 v = src[idx];
  v.x *= 2.f; v.y *= 2.f; v.z *= 2.f; v.w *= 2.f;
  ((float4*)out)[i] = v;
}

// ---------------------------------------------------------------------------
// Per (dir, b, h): features via WMMA, S = Q Kᵀ / C via WMMA, band mask + cost
// ---------------------------------------------------------------------------
__global__ void __launch_bounds__(256)
pab_attn_kernel(const float* __restrict__ xl, const float* __restrict__ xr,
                const float* __restrict__ cost,
                const float* __restrict__ qw, const float* __restrict__ qb,
                const float* __restrict__ qg, const float* __restrict__ qbeta,
                const float* __restrict__ qm, const float* __restrict__ qv,
                const float* __restrict__ kw, const float* __restrict__ kb,
                const float* __restrict__ kg, const float* __restrict__ kbeta,
                const float* __restrict__ km, const float* __restrict__ kv,
                float* __restrict__ outCost) {
  extern __shared__ float lds[];
  const int tid  = threadIdx.x;
  const int lane = tid & 31;
  const int wave = tid >> 5;
  const int m16  = lane & 15;     // M/N index within a 16-wide tile
  const int hl   = lane >> 4;     // lane half (K/M offset select)

  const int bx  = blockIdx.x;
  const int dir = bx >> 9;        // / (B*H) = 512
  const int rem = bx & 511;
  const int b   = rem >> 7;       // / H
  const int h   = rem & 127;

  // ---- Phase 0: fold BN into effective weights / bias --------------------
  for (int i = tid; i < C_*C_; i += 256) {
    int o = i >> 5, c = i & 31;
    float sq = qg[o] * rsqrtf(qv[o] + EPS_);
    float sk = kg[o] * rsqrtf(kv[o] + EPS_);
    lds[OFF_WQ + o*WSTR + c] = qw[i] * sq;
    lds[OFF_WK + o*WSTR + c] = kw[i] * sk;
  }
  if (tid < C_) {
    float sq = qg[tid] * rsqrtf(qv[tid] + EPS_);
    float sk = kg[tid] * rsqrtf(kv[tid] + EPS_);
    lds[OFF_BQ + tid] = (qb[tid] - qm[tid]) * sq + qbeta[tid];
    lds[OFF_BK + tid] = (kb[tid] - km[tid]) * sk + kbeta[tid];
  }

  // ---- Phase 1: stage X rows transposed into LDS [w][c] ------------------
  const float* xq = (dir == 0) ? xl : xr;    // Q source
  const float* xk = (dir == 0) ? xr : xl;    // K source
  const size_t rowBase = (size_t)b * (C_*H_*W_) + (size_t)h * W_;   // + c*H*W + w
  for (int i = tid; i < C_*W_; i += 256) {
    int c = i >> 8;          // / W
    int w = i & 255;
    float vq = xq[rowBase + (size_t)c * (H_*W_) + w];
    float vk = xk[rowBase + (size_t)c * (H_*W_) + w];
    lds[OFF_XQ + w*XSTR + c] = vq;
    lds[OFF_XK + w*XSTR + c] = vk;
  }
  __syncthreads();

  // ---- Phase 2: features F = X * Weffᵀ + beff  (16x16x4 f32 WMMA) --------
  // 64 tile jobs: {Q,K} x 2 N-tiles x 16 M-tiles, 8 per wave
  for (int t = 0; t < 8; ++t) {
    int job   = wave + 8*t;
    int mat   = job >> 5;          // 0 = Q, 1 = K
    int ntile = (job >> 4) & 1;
    int mtile = job & 15;
    int i0 = mtile * 16, n0 = ntile * 16;
    const float* sX = lds + (mat == 0 ? OFF_XQ : OFF_XK);
    const float* sW = lds + (mat == 0 ? OFF_WQ : OFF_WK);
    const float* sB = lds + (mat == 0 ? OFF_BQ : OFF_BK);
    float*       sF = lds + (mat == 0 ? OFF_Q  : OFF_K);

    // A lane layout (16x4 f32): row = i0+m16, k = kbase + hl*2 + {0,1}
    // B lane layout (4x16 f32): B[k][n] = Weff[n0+n][k] -> same float2 pattern
    const int aBase = (i0 + m16)*XSTR + hl*2;
    const int bBase = (n0 + m16)*WSTR + hl*2;
    v8f acc = {};
#pragma unroll
    for (int kk = 0; kk < 8; ++kk) {
      v2f a  = *(const v2f*)&sX[aBase + kk*4];
      v2f bb = *(const v2f*)&sW[bBase + kk*4];
      acc = __builtin_amdgcn_wmma_f32_16x16x4_f32(false, a, false, bb,
                                                  (short)0, acc, false, false);
    }
    float bias = sB[n0 + m16];
    // C/D layout: lane<16 -> M=v, lane>=16 -> M=8+v ; N = m16
#pragma unroll
    for (int v = 0; v < 8; ++v) {
      int m = v + hl*8;
      sF[(i0 + m)*XSTR + n0 + m16] = acc[v] + bias;
    }
  }
  __syncthreads();

  // ---- Phase 3: S = Qf Kfᵀ / C, band mask, + cost ------------------------
  const size_t plane = (size_t)(b*H_ + h) * (W_*W_);
  const size_t dstep = (size_t)B_ * H_ * W_ * W_;
  const float* costBase = cost    + (size_t)dir * dstep + plane;
  float*       outBase  = outCost + (size_t)dir * dstep + plane;

  for (int t = 0; t < 32; ++t) {           // 256 tiles / 8 waves
    int tile  = wave*32 + t;
    int itile = tile >> 4, jtile = tile & 15;
    int i0 = itile*16, j0 = jtile*16;
    int iRow = i0 + hl*8;                  // row of acc[0] for this lane
    int j    = j0 + m16;
    int dBase = (dir == 0) ? (i0 - j0) : (j0 - i0);

    if (dBase >= -15 && dBase <= (CR_ + 14)) {      // tile intersects band
      const int aBase = (i0 + m16)*XSTR + hl*2;
      const int bBase = (j0 + m16)*XSTR + hl*2;     // B[k][n] = Kf[j0+n][k]
      v8f acc = {};
#pragma unroll
      for (int kk = 0; kk < 8; ++kk) {
        v2f a  = *(const v2f*)&lds[OFF_Q + aBase + kk*4];
        v2f bb = *(const v2f*)&lds[OFF_K + bBase + kk*4];
        acc = __builtin_amdgcn_wmma_f32_16x16x4_f32(false, a, false, bb,
                                                    (short)0, acc, false, false);
      }
#pragma unroll
      for (int v = 0; v < 8; ++v) {
        int ii = iRow + v;
        int d  = (dir == 0) ? (ii - j) : (j - ii);
        float s = (d >= 0 && d < CR_) ? acc[v] * (1.0f / C_) : 0.0f;
        outBase[(size_t)ii*W_ + j] = s + costBase[(size_t)ii*W_ + j];
      }
    } else {                                         // fully masked: cost only
#pragma unroll
      for (int v = 0; v < 8; ++v) {
        int ii = iRow + v;
        outBase[(size_t)ii*W_ + j] = costBase[(size_t)ii*W_ + j];
      }
    }
  }
}

// ---------------------------------------------------------------------------
extern "C" void kernel_launch(void* const* d_in, const int* in_sizes, int n_in,
                              void* d_out, int out_size, void* d_ws, size_t ws_size,
                              hipStream_t stream) {
  (void)in_sizes; (void)n_in; (void)out_size; (void)d_ws; (void)ws_size;

  const float* xl    = (const float*)d_in[0];
  const float* xr    = (const float*)d_in[1];
  const float* cost  = (const float*)d_in[2];
  const float* qw    = (const float*)d_in[3];
  const float* qb    = (const float*)d_in[4];
  const float* qg    = (const float*)d_in[5];
  const float* qbe   = (const float*)d_in[6];
  const float* qm    = (const float*)d_in[7];
  const float* qv    = (const float*)d_in[8];
  const float* kw    = (const float*)d_in[9];
  const float* kb    = (const float*)d_in[10];
  const float* kg    = (const float*)d_in[11];
  const float* kbe   = (const float*)d_in[12];
  const float* km    = (const float*)d_in[13];
  const float* kv    = (const float*)d_in[14];
  float* out = (float*)d_out;

  const int N1 = B_ * C_ * H_ * W_;               // 4,194,304
  // outputs: [2*x_left | 2*x_right | cost_r2l | cost_l2r]
  int total4 = (2 * N1) >> 2;                     // float4 count
  pab_double_kernel<<<total4 / 256, 256, 0, stream>>>(xl, xr, out, N1);

  size_t ldsBytes = (size_t)LDS_FLOATS * sizeof(float);
  pab_attn_kernel<<<2 * B_ * H_, 256, ldsBytes, stream>>>(
      xl, xr, cost, qw, qb, qg, qbe, qm, qv, kw, kb, kg, kbe, km, kv,
      out + 2 * (size_t)N1);
}